// GCNEncoder_13511967113872
// MI455X (gfx1250) — compile-verified
//
#include <hip/hip_runtime.h>

#define N_NODES 100000
#define N_EDGES 1600000
#define IN_DIM  256
#define OUT_DIM 128

typedef __attribute__((ext_vector_type(2))) float v2f;
typedef __attribute__((ext_vector_type(4))) float v4f;
typedef __attribute__((ext_vector_type(8))) float v8f;

// ---------------------------------------------------------------- zero init
__global__ __launch_bounds__(256) void k_zero(float* __restrict__ out,
                                              float* __restrict__ deg_s,
                                              float* __restrict__ deg_d) {
  long long i = (long long)blockIdx.x * blockDim.x + threadIdx.x;
  const long long total = (long long)N_NODES * OUT_DIM;
  if (i < total) out[i] = 0.0f;
  if (i < N_NODES) { deg_s[i] = 0.0f; deg_d[i] = 0.0f; }
}

// ---------------------------------------------------------------- degrees
__global__ __launch_bounds__(256) void k_degrees(const int* __restrict__ src,
                                                 const int* __restrict__ dst,
                                                 float* __restrict__ deg_s,
                                                 float* __restrict__ deg_d) {
  int e = blockIdx.x * blockDim.x + threadIdx.x;
  if (e < N_EDGES) {
    atomicAdd(&deg_s[src[e]], 1.0f);
    atomicAdd(&deg_d[dst[e]], 1.0f);
  }
}

// ---------------------------------------------------------------- WMMA GEMM
// h = (feat * norm_s[:,None]) @ W
// Block = 256 threads = 8 waves. Each wave: 16 rows x 128 cols output strip.
// W staged through LDS in 64-row (32KB) chunks, K-pair interleaved so each
// B fragment is one aligned ds_load_b64 feeding WMMA directly.
__global__ __launch_bounds__(256) void k_gemm_wmma(const float* __restrict__ feat,
                                                   const float* __restrict__ W,
                                                   const float* __restrict__ deg_s,
                                                   float* __restrict__ h) {
  // layout: wl[(p*128 + c)*2 + 0] = W[kb+2p][c], ...*2 + 1 = W[kb+2p+1][c]
  __shared__ float wl[64 * OUT_DIM];  // 32 KB chunk of W

  const int tid  = threadIdx.x;
  const int wave = tid >> 5;
  const int lane = tid & 31;
  const int lh   = lane >> 4;   // 0: K pair {0,1}, 1: K pair {2,3}
  const int lm   = lane & 15;   // row (A) / col (B,C) within 16x16 tile

  const int rowBase = blockIdx.x * 128 + wave * 16;
  const int row     = rowBase + lm;
  const int rowc    = row < N_NODES ? row : (N_NODES - 1);  // clamp; ns=0 kills OOB
  const float* frow = feat + (size_t)rowc * IN_DIM;

  float ns = 0.0f;
  if (row < N_NODES) ns = rsqrtf(fmaxf(deg_s[row], 1.0f));

  v8f acc[8] = {};

  for (int kb = 0; kb < IN_DIM; kb += 64) {
    __syncthreads();
    // cooperative fill: 32 K-pairs x 128 cols, pair-interleaved, b128 stores
    {
      const float* Wk = W + (size_t)kb * OUT_DIM;
      #pragma unroll
      for (int i = 0; i < 8; ++i) {
        const int task = tid + i * 256;       // 0..2047
        const int p    = task >> 6;           // K-pair index 0..31
        const int c    = (task & 63) << 1;    // even column 0..126
        const v2f r0 = *(const v2f*)(Wk + (size_t)(2 * p)     * OUT_DIM + c);
        const v2f r1 = *(const v2f*)(Wk + (size_t)(2 * p + 1) * OUT_DIM + c);
        v4f q;
        q.x = r0.x; q.y = r1.x; q.z = r0.y; q.w = r1.y;
        *(v4f*)(wl + ((size_t)(p << 7) + c) * 2) = q;
      }
    }
    __syncthreads();

    // prefetch all 16 A fragments for this chunk (latency hides behind WMMAs)
    v2f afrag[16];
    #pragma unroll
    for (int j = 0; j < 16; ++j) {
      v2f a = *(const v2f*)(frow + kb + (j << 2) + 2 * lh);
      a.x *= ns;
      a.y *= ns;
      afrag[j] = a;
    }

    #pragma unroll
    for (int k0 = 0; k0 < 64; k0 += 4) {
      const int p = (k0 >> 1) + lh;                      // K-pair for this lane-half
      const float* bbase = wl + ((size_t)(p << 7) + lm) * 2;
      const v2f a = afrag[k0 >> 2];
      #pragma unroll
      for (int t = 0; t < 8; ++t) {
        const v2f b = *(const v2f*)(bbase + (t << 5));   // one ds_load_b64
        acc[t] = __builtin_amdgcn_wmma_f32_16x16x4_f32(
            /*neg_a=*/false, a, /*neg_b=*/false, b,
            /*c_mod=*/(short)0, acc[t], /*reuse_a=*/false, /*reuse_b=*/false);
      }
    }
  }

  // Store: C/D layout -> VGPR g: lanes 0-15 row=g, lanes 16-31 row=g+8; col=lm
  #pragma unroll
  for (int t = 0; t < 8; ++t) {
    #pragma unroll
    for (int g = 0; g < 8; ++g) {
      const int r = rowBase + g + 8 * lh;
      if (r < N_NODES)
        h[(size_t)r * OUT_DIM + t * 16 + lm] = acc[t][g];
    }
  }
}

// ---------------------------------------------------------------- edge scatter-add
// One 32-lane group per edge; each lane moves 4 channels (16B vector load,
// 4x global_atomic_add_f32). h and out are L2-resident (51MB each vs 192MB L2).
__global__ __launch_bounds__(256) void k_edge_agg(const int* __restrict__ src,
                                                  const int* __restrict__ dst,
                                                  const float* __restrict__ h,
                                                  float* __restrict__ out) {
  const long long gid = (long long)blockIdx.x * blockDim.x + threadIdx.x;
  const int e    = (int)(gid >> 5);
  const int lane = (int)(gid & 31);
  if (e < N_EDGES) {
    const int s = src[e];
    const int d = dst[e];
    const v4f v = *(const v4f*)(h + (size_t)s * OUT_DIM + lane * 4);
    float* o = out + (size_t)d * OUT_DIM + lane * 4;
    atomicAdd(o + 0, v.x);
    atomicAdd(o + 1, v.y);
    atomicAdd(o + 2, v.z);
    atomicAdd(o + 3, v.w);
  }
}

// ---------------------------------------------------------------- finalize
// out = agg * norm_d + b; PReLU(alpha)
__global__ __launch_bounds__(256) void k_finalize(float* __restrict__ out,
                                                  const float* __restrict__ deg_d,
                                                  const float* __restrict__ bias,
                                                  const float* __restrict__ alpha) {
  const long long i = (long long)blockIdx.x * blockDim.x + threadIdx.x;
  const long long total = (long long)N_NODES * OUT_DIM;
  if (i < total) {
    const int r = (int)(i >> 7);
    const int c = (int)(i & 127);
    const float nd = rsqrtf(fmaxf(deg_d[r], 1.0f));
    const float v  = out[i] * nd + bias[c];
    out[i] = v > 0.0f ? v : alpha[c] * v;
  }
}

// ---------------------------------------------------------------- launcher
extern "C" void kernel_launch(void* const* d_in, const int* in_sizes, int n_in,
                              void* d_out, int out_size, void* d_ws, size_t ws_size,
                              hipStream_t stream) {
  const float* feat  = (const float*)d_in[0];
  const float* W     = (const float*)d_in[1];
  const float* bias  = (const float*)d_in[2];
  const float* alpha = (const float*)d_in[3];
  const int*   src   = (const int*)d_in[4];
  const int*   dst   = (const int*)d_in[5];
  float* out = (float*)d_out;

  // workspace layout: h [N*128] | deg_s [N] | deg_d [N]
  float* h     = (float*)d_ws;
  float* deg_s = h + (size_t)N_NODES * OUT_DIM;
  float* deg_d = deg_s + N_NODES;

  const int elemBlocks = (N_NODES * OUT_DIM + 255) / 256;        // 50000
  const int edgeBlocks = (N_EDGES + 255) / 256;                  // 6250
  const int gemmBlocks = (N_NODES + 127) / 128;                  // 782
  const long long aggThreads = (long long)N_EDGES * 32;
  const int aggBlocks = (int)((aggThreads + 255) / 256);         // 200000

  k_zero<<<elemBlocks, 256, 0, stream>>>(out, deg_s, deg_d);
  k_degrees<<<edgeBlocks, 256, 0, stream>>>(src, dst, deg_s, deg_d);
  k_gemm_wmma<<<gemmBlocks, 256, 0, stream>>>(feat, W, deg_s, h);
  k_edge_agg<<<aggBlocks, 256, 0, stream>>>(src, dst, h, out);
  k_finalize<<<elemBlocks, 256, 0, stream>>>(out, deg_d, bias, alpha);
}